// Attention_80942953660799
// MI455X (gfx1250) — compile-verified
//
#include <hip/hip_runtime.h>

#define B_ 4
#define T_ 1024
#define D_ 512
#define U_ 32
#define WIN 64
#define KPAD 80        // window (79 valid j's) padded to multiple of 4
#define EPS_ 1e-7f

typedef __attribute__((ext_vector_type(2))) float v2f;
typedef __attribute__((ext_vector_type(8))) float v8f;

// ---------------------------------------------------------------------------
// Kernel 1: q = x@Wt, k = x@Wx via V_WMMA_F32_16X16X4_F32.
// grid = 256 row-tiles (16 rows each), block = 128 (4 waves).
// wave 0/1 -> Wt n-tiles 0/1 (cols 0-15 / 16-31) -> q
// wave 2/3 -> Wx n-tiles 0/1                      -> k
// ---------------------------------------------------------------------------
__global__ __launch_bounds__(128) void qk_proj_kernel(
    const float* __restrict__ x, const float* __restrict__ Wt,
    const float* __restrict__ Wx, float* __restrict__ qg,
    float* __restrict__ kg)
{
    const int mt   = blockIdx.x;          // 16-row tile of 4096 rows
    const int wave = threadIdx.x >> 5;
    const int lane = threadIdx.x & 31;
    const int g    = lane >> 4;           // lane half-group
    const int ln   = lane & 15;
    const int row0 = mt * 16;

    const float* W   = (wave < 2) ? Wt : Wx;
    float*       out = (wave < 2) ? qg : kg;
    const int    n0  = (wave & 1) * 16;

    // A-matrix 16x4 fp32 layout: lanes 0-15 hold M=0..15 K={k0,k0+1};
    // lanes 16-31 hold M=0..15 K={k0+2,k0+3}.  (2 VGPRs = float2)
    const float* arow = x + (size_t)(row0 + ln) * D_;

    v8f acc = {};
    for (int k0 = 0; k0 < D_; k0 += 4) {
        const int kk = k0 + 2 * g;
        v2f a = *(const v2f*)(arow + kk);          // contiguous, 8B aligned
        v2f b;                                     // B 4x16: K=kk,kk+1 ; N=ln
        b.x = W[(size_t)kk * U_ + n0 + ln];
        b.y = W[(size_t)(kk + 1) * U_ + n0 + ln];
        acc = __builtin_amdgcn_wmma_f32_16x16x4_f32(
            false, a, false, b, (short)0, acc, false, false);
    }
    // D 16x16 fp32: VGPR r, group g -> M = r + 8g, N = ln
#pragma unroll
    for (int r = 0; r < 8; ++r)
        out[(size_t)(row0 + r + 8 * g) * U_ + n0 + ln] = acc[r];
}

// ---------------------------------------------------------------------------
// Kernel 2: banded additive attention + v = a@x, fused per 16-row tile.
// grid = 256 tiles, block = 256 (8 waves; each wave owns 4 n-tiles of D=512).
// ---------------------------------------------------------------------------
__global__ __launch_bounds__(256) void attn_kernel(
    const float* __restrict__ x,  const float* __restrict__ qg,
    const float* __restrict__ kg, const float* __restrict__ bh,
    const float* __restrict__ Wa, const float* __restrict__ ba,
    float* __restrict__ v)
{
    __shared__ float sQ[16 * U_];       // q tile      (2 KB)
    __shared__ float sK[KPAD * U_];     // k window    (10 KB)
    __shared__ float sE[16 * KPAD];     // exp scores  (5 KB)
    __shared__ float sInv[16];          // 1/(rowsum+eps)
    __shared__ float sBh[U_], sWa[U_];

    const int tile = blockIdx.x;
    const int b    = tile / (T_ / 16);
    const int t0   = (tile % (T_ / 16)) * 16;
    const int tid  = threadIdx.x;
    const int row0 = b * T_ + t0;       // global q/v row base
    const int j0   = t0 - WIN / 2;      // window base (may be negative)

    // ---- stage q tile and k window into LDS ----
    for (int i = tid; i < 16 * U_; i += 256)
        sQ[i] = qg[(size_t)row0 * U_ + i];
    for (int i = tid; i < KPAD * U_; i += 256) {
        const int r = i >> 5, u = i & 31;
        const int jj = j0 + r;
        sK[i] = (jj >= 0 && jj < T_) ? kg[((size_t)b * T_ + jj) * U_ + u] : 0.0f;
    }
    if (tid < U_) { sBh[tid] = bh[tid]; sWa[tid] = Wa[tid]; }
    __syncthreads();

    // ---- emission scores: e = exp(tanh(q+k+bh).Wa + ba), band-masked ----
    const float ba0 = ba[0];
    for (int p = tid; p < 16 * KPAD; p += 256) {
        const int r = p / KPAD, jw = p % KPAD;
        const int t = t0 + r, jj = j0 + jw;
        const bool valid = (jj >= 0) && (jj < T_) &&
                           ((t - jj) <= WIN / 2) && ((jj - t) <= (WIN - 1) / 2);
        float acc = ba0;
#pragma unroll 8
        for (int u = 0; u < U_; ++u)
            acc += tanhf(sQ[r * U_ + u] + sK[jw * U_ + u] + sBh[u]) * sWa[u];
        sE[p] = valid ? __expf(acc) : 0.0f;
    }
    __syncthreads();

    // ---- row normalizers (scale folded into A-operand loads below) ----
    if (tid < 16) {
        float s = 0.0f;
        for (int jw = 0; jw < KPAD; ++jw) s += sE[tid * KPAD + jw];
        sInv[tid] = 1.0f / (s + EPS_);
    }
    __syncthreads();

    // ---- v[16 x 512] = a[16 x 80] @ x_window[80 x 512] via fp32 WMMA ----
    const int wave = tid >> 5, lane = tid & 31;
    const int g = lane >> 4, ln = lane & 15;
    const float invm  = sInv[ln];       // A-row scale (M = ln for both halves)
    const int   nbase = wave * 64;      // this wave's 4 n-tiles

    v8f acc0 = {}, acc1 = {}, acc2 = {}, acc3 = {};
    for (int k0 = 0; k0 < KPAD; k0 += 4) {
        const int kk = k0 + 2 * g;
        v2f a;
        a.x = sE[ln * KPAD + kk]     * invm;
        a.y = sE[ln * KPAD + kk + 1] * invm;
        // B rows kk, kk+1 of the x window; clamp OOB rows (their a == 0)
        int j1 = j0 + kk;     j1 = j1 < 0 ? 0 : (j1 >= T_ ? T_ - 1 : j1);
        int j2 = j0 + kk + 1; j2 = j2 < 0 ? 0 : (j2 >= T_ ? T_ - 1 : j2);
        const float* xr1 = x + ((size_t)b * T_ + j1) * D_ + nbase + ln;
        const float* xr2 = x + ((size_t)b * T_ + j2) * D_ + nbase + ln;
        v2f b0; b0.x = xr1[0];  b0.y = xr2[0];
        v2f b1; b1.x = xr1[16]; b1.y = xr2[16];
        v2f b2; b2.x = xr1[32]; b2.y = xr2[32];
        v2f b3; b3.x = xr1[48]; b3.y = xr2[48];
        acc0 = __builtin_amdgcn_wmma_f32_16x16x4_f32(false, a, false, b0, (short)0, acc0, false, false);
        acc1 = __builtin_amdgcn_wmma_f32_16x16x4_f32(false, a, false, b1, (short)0, acc1, false, false);
        acc2 = __builtin_amdgcn_wmma_f32_16x16x4_f32(false, a, false, b2, (short)0, acc2, false, false);
        acc3 = __builtin_amdgcn_wmma_f32_16x16x4_f32(false, a, false, b3, (short)0, acc3, false, false);
    }
#pragma unroll
    for (int r = 0; r < 8; ++r) {
        const size_t orow = ((size_t)row0 + r + 8 * g) * D_;
        v[orow + nbase + ln +  0] = acc0[r];
        v[orow + nbase + ln + 16] = acc1[r];
        v[orow + nbase + ln + 32] = acc2[r];
        v[orow + nbase + ln + 48] = acc3[r];
    }
}

// ---------------------------------------------------------------------------
extern "C" void kernel_launch(void* const* d_in, const int* in_sizes, int n_in,
                              void* d_out, int out_size, void* d_ws, size_t ws_size,
                              hipStream_t stream) {
    const float* x  = (const float*)d_in[0];
    const float* Wt = (const float*)d_in[1];
    const float* Wx = (const float*)d_in[2];
    const float* bh = (const float*)d_in[3];
    const float* Wa = (const float*)d_in[4];
    const float* ba = (const float*)d_in[5];
    float* out = (float*)d_out;

    float* qg = (float*)d_ws;                       // [4096 x 32]
    float* kg = qg + (size_t)B_ * T_ * U_;          // [4096 x 32]  (1 MB total)

    qk_proj_kernel<<<(B_ * T_) / 16, 128, 0, stream>>>(x, Wt, Wx, qg, kg);
    attn_kernel  <<<(B_ * T_) / 16, 256, 0, stream>>>(x, qg, kg, bh, Wa, ba, out);
}